// KAEM_layer_23845658428265
// MI455X (gfx1250) — compile-verified
//
#include <hip/hip_runtime.h>
#include <math.h>

#define BB 16
#define CC 256
#define HH 64
#define WWID 64
#define HW (HH * WWID)
#define INNER 512
#define GC 32
#define ID_C (CC - 3 * GC) /* 160 */
#define EPSV 0.0005f

typedef __attribute__((ext_vector_type(2))) float v2f;
typedef __attribute__((ext_vector_type(4))) float v4f;
typedef __attribute__((ext_vector_type(8))) float v8f;

// ---------------------------------------------------------------------------
// Kernel 0: m = sigmoid(mask), area[b] = sum(m) + eps
// ---------------------------------------------------------------------------
__global__ void k_mask(const float* __restrict__ mask,
                       float* __restrict__ m,
                       float* __restrict__ area) {
  __shared__ float red[256];
  const int b = blockIdx.x;
  float s = 0.f;
  for (int p = threadIdx.x; p < HW; p += 256) {
    float v = 1.f / (1.f + expf(-mask[b * HW + p]));
    m[b * HW + p] = v;
    s += v;
  }
  red[threadIdx.x] = s;
  __syncthreads();
  for (int off = 128; off > 0; off >>= 1) {
    if ((int)threadIdx.x < off) red[threadIdx.x] += red[threadIdx.x + off];
    __syncthreads();
  }
  if (threadIdx.x == 0) area[b] = red[0] + EPSV;
}

// ---------------------------------------------------------------------------
// Kernel 1: inception depthwise conv fused with masked average pool.
// One block per (channel, batch). Conv channels stage the 16KB plane in LDS.
// xpool[b*CC + c] = sum_hw conv(x)[c,hw] * m[b,hw] / area[b]
// ---------------------------------------------------------------------------
__global__ void k_incpool(const float* __restrict__ x,
                          const float* __restrict__ m,
                          const float* __restrict__ area,
                          const float* __restrict__ w_hw, const float* __restrict__ b_hw,
                          const float* __restrict__ w_w,  const float* __restrict__ b_w,
                          const float* __restrict__ w_h,  const float* __restrict__ b_h,
                          float* __restrict__ xpool) {
  __shared__ float plane[HW];
  __shared__ float red[256];
  const int c = blockIdx.x;
  const int b = blockIdx.y;
  const float* xp = x + ((size_t)b * CC + c) * HW;
  const float* mp = m + (size_t)b * HW;
  float s = 0.f;

  if (c < ID_C) {
    // identity branch
    for (int p = threadIdx.x; p < HW; p += 256) s += xp[p] * mp[p];
  } else {
    // stage full channel plane into LDS (16 KB), then run depthwise conv
    const v4f* x4 = (const v4f*)xp;
    v4f* p4 = (v4f*)plane;
    for (int i = threadIdx.x; i < HW / 4; i += 256) p4[i] = x4[i];
    __syncthreads();

    if (c < ID_C + GC) {
      // 3x3, pad (1,1)
      const int g = c - ID_C;
      float kw9[9];
#pragma unroll
      for (int i = 0; i < 9; ++i) kw9[i] = w_hw[g * 9 + i];
      const float bias = b_hw[g];
      for (int p = threadIdx.x; p < HW; p += 256) {
        const int h = p >> 6, w = p & 63;
        float acc = bias;
#pragma unroll
        for (int kh = 0; kh < 3; ++kh) {
          const int hh = h + kh - 1;
          if ((unsigned)hh < 64u) {
#pragma unroll
            for (int kw = 0; kw < 3; ++kw) {
              const int ww = w + kw - 1;
              if ((unsigned)ww < 64u) acc += plane[(hh << 6) + ww] * kw9[kh * 3 + kw];
            }
          }
        }
        s += acc * mp[p];
      }
    } else if (c < ID_C + 2 * GC) {
      // 1x11, pad (0,5)
      const int g = c - (ID_C + GC);
      float k11[11];
#pragma unroll
      for (int i = 0; i < 11; ++i) k11[i] = w_w[g * 11 + i];
      const float bias = b_w[g];
      for (int p = threadIdx.x; p < HW; p += 256) {
        const int h = p >> 6, w = p & 63;
        float acc = bias;
#pragma unroll
        for (int kw = 0; kw < 11; ++kw) {
          const int ww = w + kw - 5;
          if ((unsigned)ww < 64u) acc += plane[(h << 6) + ww] * k11[kw];
        }
        s += acc * mp[p];
      }
    } else {
      // 11x1, pad (5,0)
      const int g = c - (ID_C + 2 * GC);
      float k11[11];
#pragma unroll
      for (int i = 0; i < 11; ++i) k11[i] = w_h[g * 11 + i];
      const float bias = b_h[g];
      for (int p = threadIdx.x; p < HW; p += 256) {
        const int h = p >> 6, w = p & 63;
        float acc = bias;
#pragma unroll
        for (int kh = 0; kh < 11; ++kh) {
          const int hh = h + kh - 5;
          if ((unsigned)hh < 64u) acc += plane[(hh << 6) + w] * k11[kh];
        }
        s += acc * mp[p];
      }
    }
  }

  red[threadIdx.x] = s;
  __syncthreads();
  for (int off = 128; off > 0; off >>= 1) {
    if ((int)threadIdx.x < off) red[threadIdx.x] += red[threadIdx.x + off];
    __syncthreads();
  }
  if (threadIdx.x == 0) xpool[b * CC + c] = red[0] / area[b];
}

// ---------------------------------------------------------------------------
// Kernel 2a: vf[o,b] = Wv[o,:] . xpool[b,:]   (M=512, N=16, K=256)
// fp32 WMMA 16x16x4. One wave per 16-row M tile; 32 waves total.
// A 16x4 layout: lanes 0-15 -> K={0,1} in vgpr{0,1}; lanes 16-31 -> K={2,3}.
// B 4x16 mirror: lanes 0-15 hold N=0..15 for K={0,1}; lanes 16-31 for K={2,3}.
// vf stored [o, b] (M-major).
// ---------------------------------------------------------------------------
__global__ void k_gemm_v(const float* __restrict__ w_qkv,
                         const float* __restrict__ xpool,
                         float* __restrict__ vf) {
  const int wave = (int)((blockIdx.x * blockDim.x + threadIdx.x) >> 5); // 0..31
  const int lane = threadIdx.x & 31;
  const int m0 = wave * 16;
  const int row = lane & 15;            // M row for A, N col for B
  const int khalf = (lane >> 4) * 2;    // 0 or 2
  const float* A = w_qkv + (size_t)(2 * INNER) * CC; // V block rows [1024,1536)

  v8f acc = {};
  for (int kk = 0; kk < CC / 4; ++kk) {
    const int kbase = kk * 4 + khalf;
    v2f a = *(const v2f*)(A + (size_t)(m0 + row) * CC + kbase);
    v2f bf;
    bf.x = xpool[row * CC + kbase];
    bf.y = xpool[row * CC + kbase + 1];
    acc = __builtin_amdgcn_wmma_f32_16x16x4_f32(false, a, false, bf,
                                                (short)0, acc, false, false);
  }
  // D layout: vgpr v -> M = v (lanes 0-15) or 8+v (lanes 16-31); N = lane&15
  const int nCol = lane & 15;
  const int mAdd = (lane >> 4) * 8;
#pragma unroll
  for (int v = 0; v < 8; ++v)
    vf[(m0 + v + mAdd) * BB + nCol] = acc[v];
}

// ---------------------------------------------------------------------------
// Kernel 2b: final[b,c] = w_out[c,:] . vf[:,b] + b_out[c]  (M=256,N=16,K=512)
// ---------------------------------------------------------------------------
__global__ void k_gemm_out(const float* __restrict__ w_out,
                           const float* __restrict__ b_out,
                           const float* __restrict__ vf,
                           float* __restrict__ fin) {
  const int wave = (int)((blockIdx.x * blockDim.x + threadIdx.x) >> 5); // 0..15
  const int lane = threadIdx.x & 31;
  const int m0 = wave * 16;
  const int row = lane & 15;
  const int khalf = (lane >> 4) * 2;

  v8f acc = {};
  for (int kk = 0; kk < INNER / 4; ++kk) {
    const int kbase = kk * 4 + khalf;
    v2f a = *(const v2f*)(w_out + (size_t)(m0 + row) * INNER + kbase);
    v2f bf;
    bf.x = vf[kbase * BB + row];
    bf.y = vf[(kbase + 1) * BB + row];
    acc = __builtin_amdgcn_wmma_f32_16x16x4_f32(false, a, false, bf,
                                                (short)0, acc, false, false);
  }
  const int nCol = lane & 15;
  const int mAdd = (lane >> 4) * 8;
#pragma unroll
  for (int v = 0; v < 8; ++v) {
    const int M = m0 + v + mAdd;
    fin[nCol * CC + M] = acc[v] + b_out[M];
  }
}

// ---------------------------------------------------------------------------
// Kernel 3: broadcast final[b,:] over all 4096 pixels. Pure bandwidth:
// 67 MB of B128 non-temporal stores (output never re-read -> bypass L2 rinse).
// ---------------------------------------------------------------------------
__global__ void k_broadcast(const float* __restrict__ fin,
                            float* __restrict__ out) {
  const size_t idx = (size_t)blockIdx.x * blockDim.x + threadIdx.x; // float4 units
  const int c4 = (int)(idx & 63);        // c/4
  const int b = (int)(idx >> 18);        // 4096*64 float4 per batch
  v4f val = ((const v4f*)fin)[b * 64 + c4];
  __builtin_nontemporal_store(val, ((v4f*)out) + idx);
}

// ---------------------------------------------------------------------------
extern "C" void kernel_launch(void* const* d_in, const int* in_sizes, int n_in,
                              void* d_out, int out_size, void* d_ws, size_t ws_size,
                              hipStream_t stream) {
  const float* x     = (const float*)d_in[0];
  const float* mask  = (const float*)d_in[1];
  const float* w_hw  = (const float*)d_in[2];
  const float* b_hw  = (const float*)d_in[3];
  const float* w_w   = (const float*)d_in[4];
  const float* b_w   = (const float*)d_in[5];
  const float* w_h   = (const float*)d_in[6];
  const float* b_h   = (const float*)d_in[7];
  const float* w_qkv = (const float*)d_in[8];
  const float* w_out = (const float*)d_in[9];
  const float* b_out = (const float*)d_in[10];
  float* out = (float*)d_out;

  float* ws    = (float*)d_ws;
  float* m     = ws;                // 16*4096
  float* area  = m + BB * HW;       // 16
  float* xpool = area + BB;         // 16*256
  float* vf    = xpool + BB * CC;   // 512*16
  float* fin   = vf + INNER * BB;   // 16*256

  k_mask<<<BB, 256, 0, stream>>>(mask, m, area);
  k_incpool<<<dim3(CC, BB), 256, 0, stream>>>(x, m, area, w_hw, b_hw,
                                              w_w, b_w, w_h, b_h, xpool);
  k_gemm_v<<<8, 128, 0, stream>>>(w_qkv, xpool, vf);      // 32 waves, all full
  k_gemm_out<<<4, 128, 0, stream>>>(w_out, b_out, vf, fin); // 16 waves, all full
  k_broadcast<<<(BB * HW * CC / 4) / 256, 256, 0, stream>>>(fin, out);
}